// Spline1DInterpolant_51625506898528
// MI455X (gfx1250) — compile-verified
//
#include <hip/hip_runtime.h>
#include <hip/hip_bf16.h>

typedef __attribute__((ext_vector_type(2))) float v2f;
typedef __attribute__((ext_vector_type(8))) float v8f;

// One wave32 handles a tile of 16 queries.
//   A (16x4 f32): A[q][k] = c[j_q + k]   (gathered spline coeffs)
//   B (4x16 f32): B[k][q] = w_k(f_q)     (cubic B-spline weights)
//   D = A x B ; diagonal D[q][q] = interpolant value for query q.
// ISA VGPR layouts (CDNA5 ISA):
//   A: lanes 0-15 -> M=lane, VGPR0=K0, VGPR1=K1 ; lanes 16-31 -> M=lane-16, K2/K3
//   B: lanes 0-15 -> N=lane, VGPR0=row0, VGPR1=row1 ; lanes 16-31 -> rows 2/3
//   D: lanes 0-15 -> rows 0-7 (VGPR=r), lanes 16-31 -> rows 8-15
//      => diag q<8 in lane q VGPR q ; diag q>=8 in lane q+16 VGPR q-8.
__global__ __launch_bounds__(256)
void Spline1DInterpolant_kernel(const float* __restrict__ x,
                                const float* __restrict__ a,
                                const float* __restrict__ b,
                                const float* __restrict__ n,
                                const float* __restrict__ c,
                                float* __restrict__ out,
                                int nquery, int ncoef)
{
    const int tid  = blockIdx.x * blockDim.x + threadIdx.x;
    const int lane = threadIdx.x & 31;
    const int tile = tid >> 5;          // one tile of 16 queries per wave
    const int q    = lane & 15;         // query slot (column / A-row)
    const int half = lane >> 4;         // 0 -> K={0,1}, 1 -> K={2,3}
    const int base = tile * 16;

    const float a0 = a[0];
    const float h  = (b[0] - a0) / n[0];   // same op order as reference

    // All 32 lanes need query q's (j, f); lanes 16-31 just reload the same x.
    const float xq = x[base + q];
    const float s  = (xq - a0) / h;
    float jf = floorf(s);
    jf = fminf(fmaxf(jf, 0.0f), (float)(ncoef - 4));  // keep j..j+3 in range
    const int   j = (int)jf;
    const float f = s - jf;

    // A-matrix fragment: two gathered coefficients (16 KB c -> L1/WGP$ hits).
    v2f A;
    A.x = c[j + 2 * half];
    A.y = c[j + 2 * half + 1];

    // B-matrix fragment: cubic B-spline weights for this lane's K rows.
    const float omf = 1.0f - f;
    const float w0 = omf * omf * omf;                  // t = f+1 (outer)
    const float w1 = 4.0f + f * f * (3.0f * f - 6.0f); // t = f   (inner)
    const float w2 = 4.0f + omf * omf * (3.0f * omf - 6.0f); // t = 1-f
    const float w3 = f * f * f;                        // t = 2-f (outer)
    v2f Bv;
    Bv.x = half ? w2 : w0;
    Bv.y = half ? w3 : w1;

    v8f Cv = {};
    // (neg_a, A, neg_b, B, c_mod, C, reuse_a, reuse_b)
    v8f D = __builtin_amdgcn_wmma_f32_16x16x4_f32(
        false, A, false, Bv, (short)0, Cv, false, false);

    // Diagonal extraction: lanes 0-7 (VGPR=lane) and lanes 24-31 (VGPR=lane-24).
    const bool active = (lane < 8) || (lane >= 24);
    const int  idx    = (lane < 8) ? lane : (lane - 24);
    float dv = D[0];
    dv = (idx == 1) ? D[1] : dv;
    dv = (idx == 2) ? D[2] : dv;
    dv = (idx == 3) ? D[3] : dv;
    dv = (idx == 4) ? D[4] : dv;
    dv = (idx == 5) ? D[5] : dv;
    dv = (idx == 6) ? D[6] : dv;
    dv = (idx == 7) ? D[7] : dv;

    const int qout = (lane < 8) ? lane : (lane - 16);
    if (active && (base + qout) < nquery)
        out[base + qout] = dv;
}

extern "C" void kernel_launch(void* const* d_in, const int* in_sizes, int n_in,
                              void* d_out, int out_size, void* d_ws, size_t ws_size,
                              hipStream_t stream) {
    (void)n_in; (void)d_ws; (void)ws_size; (void)out_size;
    const float* x = (const float*)d_in[0];
    const float* a = (const float*)d_in[1];
    const float* b = (const float*)d_in[2];
    const float* n = (const float*)d_in[3];
    const float* c = (const float*)d_in[4];
    float* out = (float*)d_out;

    const int nquery = in_sizes[0];   // 16384
    const int ncoef  = in_sizes[4];   // 4096

    // 16 queries per wave32 -> nquery/16 waves; 256 threads = 8 waves/block.
    const int waves  = (nquery + 15) / 16;
    const int blocks = (waves + 7) / 8;
    Spline1DInterpolant_kernel<<<blocks, 256, 0, stream>>>(
        x, a, b, n, c, out, nquery, ncoef);
}